// ChamferDistance_28767690949109
// MI455X (gfx1250) — compile-verified
//
#include <hip/hip_runtime.h>
#include <hip/hip_bf16.h>

typedef __attribute__((ext_vector_type(2))) float v2f;
typedef __attribute__((ext_vector_type(8))) float v8f;

#define BATCH 4
#define NPTS  8192   // N == M == 8192

// One wave (32 lanes) owns a strip of 16 "row" points (X) and scans all "col"
// points (Y) in 16-wide tiles. For each tile:
//   D = A(16x4) * B(4x16)  via v_wmma_f32_16x16x4_f32   (K=3 padded to 4)
//   P = rx + ry - 2*D      (squared distance)
// Running min over columns kept in registers; cross-lane butterfly at the end.
__global__ __launch_bounds__(256) void chamfer_dir_kernel(
    const float* __restrict__ X,   // rows  [BATCH, NPTS, 3]
    const float* __restrict__ Y,   // cols  [BATCH, NPTS, 3]
    float* __restrict__ acc)       // single-float accumulator (sum of row-mins)
{
  const int lane = threadIdx.x & 31;
  const int wave = (blockIdx.x * blockDim.x + threadIdx.x) >> 5;
  const int stripsPerBatch = NPTS / 16;           // 512
  const int b  = wave / stripsPerBatch;
  const int r0 = (wave % stripsPerBatch) << 4;

  const int half = lane >> 4;                     // 0: K=0,1 / rows 0-7 of D
  const int l    = lane & 15;                     // 1: K=2,3 / rows 8-15 of D

  const float* Xb = X + (size_t)b * NPTS * 3;
  const float* Yb = Y + (size_t)b * NPTS * 3;

  // ---- A tile: 16 rows of X, fp32 16x4 layout (K padded to 4 with 0) ----
  const float* xp = Xb + (size_t)(r0 + l) * 3;
  const float g0 = xp[0], g1 = xp[1], g2 = xp[2];
  v2f a;
  a.x = half ? g2 : g0;
  a.y = half ? 0.0f : g1;

  // rx for this lane's row (same value in lanes l and l+16)
  const float rx_lane = g0 * g0 + g1 * g1 + g2 * g2;
  // broadcast rx for the 8 D-rows this half owns: row = i + 8*half
  float rxv[8];
#pragma unroll
  for (int i = 0; i < 8; ++i)
    rxv[i] = __shfl(rx_lane, i + 8 * half, 32);

  float rm[8];
#pragma unroll
  for (int i = 0; i < 8; ++i) rm[i] = 3.0e38f;

  for (int c0 = 0; c0 < NPTS; c0 += 16) {
    // ---- B tile: 16 cols of Y, fp32 4x16 layout (lane = column) ----
    const float* yp = Yb + (size_t)(c0 + l) * 3;
    const float p0 = yp[0], p1 = yp[1], p2 = yp[2];
    v2f bm;
    bm.x = half ? p2 : p0;
    bm.y = half ? 0.0f : p1;
    const float ry = p0 * p0 + p1 * p1 + p2 * p2;  // ry of this lane's column

    v8f cz = {0.f, 0.f, 0.f, 0.f, 0.f, 0.f, 0.f, 0.f};
    // D = A x B + 0  -> v_wmma_f32_16x16x4_f32
    v8f d = __builtin_amdgcn_wmma_f32_16x16x4_f32(
        false, a, false, bm, (short)0, cz, false, false);

#pragma unroll
    for (int i = 0; i < 8; ++i) {
      // squared distance: rx[row] + ry[col] - 2 * <x,y>
      const float P = fmaf(-2.0f, d[i], rxv[i] + ry);
      rm[i] = fminf(rm[i], P);
    }
  }

  // Min across the 16 lanes of each half (butterfly -> all lanes hold min).
#pragma unroll
  for (int i = 0; i < 8; ++i) {
    float v = rm[i];
    v = fminf(v, __shfl_xor(v, 1, 32));
    v = fminf(v, __shfl_xor(v, 2, 32));
    v = fminf(v, __shfl_xor(v, 4, 32));
    v = fminf(v, __shfl_xor(v, 8, 32));
    rm[i] = v;
  }

  // Sum the 8 row-mins of this half, combine halves, one atomic per wave.
  float s = 0.0f;
#pragma unroll
  for (int i = 0; i < 8; ++i) s += rm[i];
  s += __shfl_xor(s, 16, 32);

  if (lane == 0) atomicAdd(acc, s);
}

__global__ void chamfer_zero_kernel(float* acc) {
  if (threadIdx.x == 0 && blockIdx.x == 0) acc[0] = 0.0f;
}

__global__ void chamfer_finish_kernel(const float* __restrict__ acc,
                                      float* __restrict__ out) {
  // loss = sum(mins1)/M + sum(mins2)/N, with M == N == 8192
  out[0] = acc[0] * (1.0f / (float)NPTS);
}

extern "C" void kernel_launch(void* const* d_in, const int* in_sizes, int n_in,
                              void* d_out, int out_size, void* d_ws, size_t ws_size,
                              hipStream_t stream) {
  const float* preds = (const float*)d_in[0];  // [4, 8192, 3]
  const float* gts   = (const float*)d_in[1];  // [4, 8192, 3]
  float* out = (float*)d_out;
  float* acc = (float*)d_ws;

  chamfer_zero_kernel<<<1, 32, 0, stream>>>(acc);

  const int strips = BATCH * (NPTS / 16);      // 2048 waves per direction
  const int wavesPerBlock = 8;                 // 256 threads = 8 wave32
  dim3 grid(strips / wavesPerBlock);
  dim3 block(wavesPerBlock * 32);

  // mins2: rows = gts, min over preds; summed /8192
  chamfer_dir_kernel<<<grid, block, 0, stream>>>(gts, preds, acc);
  // mins1: rows = preds, min over gts; summed /8192
  chamfer_dir_kernel<<<grid, block, 0, stream>>>(preds, gts, acc);

  chamfer_finish_kernel<<<1, 1, 0, stream>>>(acc, out);
}